// GCN_1614907703893
// MI455X (gfx1250) — compile-verified
//
#include <hip/hip_runtime.h>

typedef __attribute__((ext_vector_type(2))) float v2f;
typedef __attribute__((ext_vector_type(8))) float v8f;

// ---------------------------------------------------------------- utilities
__global__ __launch_bounds__(256) void zero_f32(float* __restrict__ p, long long n) {
    long long i = (long long)blockIdx.x * blockDim.x + threadIdx.x;
    if (i < n) p[i] = 0.0f;
}

__global__ __launch_bounds__(256) void deg_count(const int* __restrict__ dst,
                                                 float* __restrict__ deg, int nE) {
    int e = blockIdx.x * blockDim.x + threadIdx.x;
    if (e < nE) atomicAdd(&deg[dst[e]], 1.0f);
}

__global__ __launch_bounds__(256) void deg_to_dinv(float* __restrict__ deg, int n) {
    int i = blockIdx.x * blockDim.x + threadIdx.x;
    if (i < n) deg[i] = rsqrtf(deg[i] + 1.0f);  // +1 = self loop; always > 0
}

// ---------------------------------------------------------------- WMMA GEMMs
// h1 = X(100k x 256) @ W1(256 x 16); one wave computes a 16x16 output tile,
// looping K in steps of 4 with V_WMMA_F32_16X16X4_F32 (fp32 in/out).
__global__ __launch_bounds__(256) void gemm_x_w1(const float* __restrict__ X,
                                                 const float* __restrict__ W,
                                                 float* __restrict__ H, int n) {
    const int lane = threadIdx.x & 31;
    const int tile = blockIdx.x * 8 + (threadIdx.x >> 5);
    const int row0 = tile * 16;
    if (row0 >= n) return;                       // wave-uniform (EXEC stays all-1)
    int m = row0 + (lane & 15);
    if (m >= n) m = n - 1;                       // clamp reads; stores are guarded
    const int kh  = (lane >> 4) << 1;            // 0 for lanes 0-15, 2 for 16-31
    const int col = lane & 15;
    const float* xr = X + (size_t)m * 256;

    v8f acc = {};
#pragma unroll 4
    for (int k0 = 0; k0 < 256; k0 += 4) {
        v2f a, b;
        a.x = xr[k0 + kh];
        a.y = xr[k0 + kh + 1];
        b.x = W[(size_t)(k0 + kh) * 16 + col];
        b.y = W[(size_t)(k0 + kh + 1) * 16 + col];
        acc = __builtin_amdgcn_wmma_f32_16x16x4_f32(false, a, false, b,
                                                    (short)0, acc, false, false);
    }
    // D layout: VGPR v holds row (v + 8*(lane>=16)), col = lane&15
    const int ra = (lane >> 4) << 3;
    if (row0 + 16 <= n) {                        // wave-uniform fast path: no guards
#pragma unroll
        for (int v = 0; v < 8; ++v)
            H[(size_t)(row0 + ra + v) * 16 + col] = acc[v];
    } else {
#pragma unroll
        for (int v = 0; v < 8; ++v) {
            int r = row0 + ra + v;
            if (r < n) H[(size_t)r * 16 + col] = acc[v];
        }
    }
}

// h2 = Z(100k x 16) @ W2(16 x 64); 4 accumulators cover the 64 columns.
__global__ __launch_bounds__(256) void gemm_z_w2(const float* __restrict__ Z,
                                                 const float* __restrict__ W,
                                                 float* __restrict__ H, int n) {
    const int lane = threadIdx.x & 31;
    const int tile = blockIdx.x * 8 + (threadIdx.x >> 5);
    const int row0 = tile * 16;
    if (row0 >= n) return;
    int m = row0 + (lane & 15);
    if (m >= n) m = n - 1;
    const int kh  = (lane >> 4) << 1;
    const int col = lane & 15;
    const float* zr = Z + (size_t)m * 16;

    v8f acc0 = {}, acc1 = {}, acc2 = {}, acc3 = {};
#pragma unroll
    for (int k0 = 0; k0 < 16; k0 += 4) {
        v2f a, b;
        a.x = zr[k0 + kh];
        a.y = zr[k0 + kh + 1];
        const float* w0 = W + (size_t)(k0 + kh) * 64;
        const float* w1 = w0 + 64;
        b.x = w0[col];      b.y = w1[col];
        acc0 = __builtin_amdgcn_wmma_f32_16x16x4_f32(false, a, false, b, (short)0, acc0, false, false);
        b.x = w0[col + 16]; b.y = w1[col + 16];
        acc1 = __builtin_amdgcn_wmma_f32_16x16x4_f32(false, a, false, b, (short)0, acc1, false, false);
        b.x = w0[col + 32]; b.y = w1[col + 32];
        acc2 = __builtin_amdgcn_wmma_f32_16x16x4_f32(false, a, false, b, (short)0, acc2, false, false);
        b.x = w0[col + 48]; b.y = w1[col + 48];
        acc3 = __builtin_amdgcn_wmma_f32_16x16x4_f32(false, a, false, b, (short)0, acc3, false, false);
    }
    const int ra = (lane >> 4) << 3;
    if (row0 + 16 <= n) {                        // wave-uniform fast path
#pragma unroll
        for (int v = 0; v < 8; ++v) {
            float* hp = H + (size_t)(row0 + ra + v) * 64 + col;
            hp[0]  = acc0[v];
            hp[16] = acc1[v];
            hp[32] = acc2[v];
            hp[48] = acc3[v];
        }
    } else {
#pragma unroll
        for (int v = 0; v < 8; ++v) {
            int r = row0 + ra + v;
            if (r < n) {
                float* hp = H + (size_t)r * 64 + col;
                hp[0]  = acc0[v];
                hp[16] = acc1[v];
                hp[32] = acc2[v];
                hp[48] = acc3[v];
            }
        }
    }
}

// ---------------------------------------------------------------- aggregation
// One thread per (edge, 16-feature chunk): gather h[src]*norm, atomic-add to out[dst].
template <int F>
__global__ __launch_bounds__(256) void scatter_add(const int* __restrict__ src,
                                                   const int* __restrict__ dst,
                                                   const float* __restrict__ dinv,
                                                   const float* __restrict__ H,
                                                   float* __restrict__ OUT, int nE) {
    constexpr int CH = F / 16;
    long long idx = (long long)blockIdx.x * blockDim.x + threadIdx.x;
    long long e = idx / CH;
    int c = (int)(idx % CH);
    if (e >= nE) return;
    int s = src[e], d = dst[e];
    float nrm = dinv[s] * dinv[d];
    const float* hp = H + (size_t)s * F + c * 16;
    float* op = OUT + (size_t)d * F + c * 16;
#pragma unroll
    for (int f = 0; f < 16; ++f)
        atomicAdd(op + f, hp[f] * nrm);
}

// out = agg + dinv^2 * h (self loop) + bias, optional ReLU; in place over agg.
template <int F, bool RELU>
__global__ __launch_bounds__(256) void epilogue(float* __restrict__ AGG,
                                                const float* __restrict__ H,
                                                const float* __restrict__ dinv,
                                                const float* __restrict__ bias, int n) {
    long long idx = (long long)blockIdx.x * blockDim.x + threadIdx.x;
    if (idx >= (long long)n * F) return;
    int i = (int)(idx / F);
    int f = (int)(idx % F);
    float di = dinv[i];
    float v = AGG[idx] + di * di * H[idx] + bias[f];
    if (RELU) v = fmaxf(v, 0.0f);
    AGG[idx] = v;
}

// ---------------------------------------------------------------- edge scoring
__global__ __launch_bounds__(256) void edge_score(const int* __restrict__ psrc,
                                                  const int* __restrict__ pdst,
                                                  const int* __restrict__ nsrc,
                                                  const int* __restrict__ ndst,
                                                  const float* __restrict__ Z,
                                                  float* __restrict__ out, int nT) {
    int j = blockIdx.x * blockDim.x + threadIdx.x;
    if (j >= 2 * nT) return;
    int s, d;
    if (j < nT) { s = psrc[j]; d = pdst[j]; }
    else        { s = nsrc[j - nT]; d = ndst[j - nT]; }
    const float4* zs = (const float4*)(Z + (size_t)s * 64);
    const float4* zd = (const float4*)(Z + (size_t)d * 64);
    float acc = 0.0f;
#pragma unroll
    for (int k = 0; k < 16; ++k) {
        float4 a = zs[k], b = zd[k];
        acc = fmaf(a.x, b.x, acc);
        acc = fmaf(a.y, b.y, acc);
        acc = fmaf(a.z, b.z, acc);
        acc = fmaf(a.w, b.w, acc);
    }
    out[j] = acc;
}

// ---------------------------------------------------------------- launch
extern "C" void kernel_launch(void* const* d_in, const int* in_sizes, int n_in,
                              void* d_out, int out_size, void* d_ws, size_t ws_size,
                              hipStream_t stream) {
    const float* x   = (const float*)d_in[0];
    const int*   tei = (const int*)d_in[1];   // (2, E): row0 = src, row1 = dst
    const int*   tpo = (const int*)d_in[2];   // (2, E_TEST)
    const int*   tne = (const int*)d_in[3];   // (2, E_TEST)
    const float* W1  = (const float*)d_in[4];
    const float* b1  = (const float*)d_in[5];
    const float* W2  = (const float*)d_in[6];
    const float* b2  = (const float*)d_in[7];

    const int N  = in_sizes[0] / 256;
    const int E  = in_sizes[1] / 2;
    const int ET = in_sizes[2] / 2;

    float* ws   = (float*)d_ws;
    float* dinv = ws;                          // N
    float* h1   = ws + (size_t)N;              // 16N
    float* agg1 = ws + (size_t)17 * N;         // 16N (becomes z1 in place)
    float* h2   = ws + (size_t)33 * N;         // 64N
    float* agg2 = ws + (size_t)97 * N;         // 64N (becomes z2 in place)

    const int B = 256;
    auto cdiv = [](long long a, long long b) { return (int)((a + b - 1) / b); };

    // zero the accumulation buffers every call (ws is poisoned / stale)
    zero_f32<<<cdiv(N, B), B, 0, stream>>>(dinv, N);
    zero_f32<<<cdiv((long long)16 * N, B), B, 0, stream>>>(agg1, (long long)16 * N);
    zero_f32<<<cdiv((long long)64 * N, B), B, 0, stream>>>(agg2, (long long)64 * N);

    // degree -> dinv
    deg_count<<<cdiv(E, B), B, 0, stream>>>(tei + E, dinv, E);
    deg_to_dinv<<<cdiv(N, B), B, 0, stream>>>(dinv, N);

    const int tiles = (N + 15) / 16;

    // layer 1
    gemm_x_w1<<<cdiv(tiles, 8), B, 0, stream>>>(x, W1, h1, N);
    scatter_add<16><<<cdiv(E, B), B, 0, stream>>>(tei, tei + E, dinv, h1, agg1, E);
    epilogue<16, true><<<cdiv((long long)16 * N, B), B, 0, stream>>>(agg1, h1, dinv, b1, N);

    // layer 2
    gemm_z_w2<<<cdiv(tiles, 8), B, 0, stream>>>(agg1, W2, h2, N);
    scatter_add<64><<<cdiv((long long)4 * E, B), B, 0, stream>>>(tei, tei + E, dinv, h2, agg2, E);
    epilogue<64, false><<<cdiv((long long)64 * N, B), B, 0, stream>>>(agg2, h2, dinv, b2, N);

    // link-prediction logits (pos first, then neg)
    edge_score<<<cdiv(2 * ET, B), B, 0, stream>>>(tpo, tpo + ET, tne, tne + ET,
                                                  agg2, (float*)d_out, ET);
}